// QuantumConv_12687333392419
// MI455X (gfx1250) — compile-verified
//
#include <hip/hip_runtime.h>

#define NQ 9
#define HW 28
#define NOUT 26
#define NPATCH (16 * NOUT * NOUT)   // 10816 = 1352 blocks * 8 waves

typedef __attribute__((ext_vector_type(2))) float v2f;
typedef __attribute__((ext_vector_type(8))) float v8f;

struct cplx { float re, im; };
__device__ __forceinline__ cplx cmul(cplx a, cplx b) {
    return { a.re * b.re - a.im * b.im, a.re * b.im + a.im * b.re };
}
__device__ __forceinline__ cplx cadd(cplx a, cplx b) { return { a.re + b.re, a.im + b.im }; }

struct gate2 { cplx g00, g01, g10, g11; };

__device__ __forceinline__ cplx gpick(const gate2& g, int rb, int cb) {
    cplx lo = cb ? g.g01 : g.g00;
    cplx hi = cb ? g.g11 : g.g10;
    return rb ? hi : lo;
}

__device__ __forceinline__ gate2 mk_rx(float c, float s) {
    return { {c, 0.f}, {0.f, -s}, {0.f, -s}, {c, 0.f} };
}

// D = A(16x4) * B(4x16) + C, f32, full precision
__device__ __forceinline__ v8f wmma4(v2f a, v2f b, v8f c) {
    return __builtin_amdgcn_wmma_f32_16x16x4_f32(false, a, false, b, (short)0, c, false, false);
}

// Apply 2x2 complex gate on flat amplitude bit B.
// State: 512 amps = 16x32 matrix in WMMA C/D layout, duplicated for blocks X (cols 0-15)
// and Y (cols 16-31). Bits 0-2: register pairs. Bit 3: shfl_xor 16. Bits 4-7: shfl_xor 1/2/4/8.
// Bit 8: X/Y register pairs.
template<int B>
__device__ __forceinline__ void apply1q(float* Xre, float* Xim, float* Yre, float* Yim,
                                        gate2 g, unsigned lane)
{
    if constexpr (B < 3) {
        constexpr int m = 1 << B;
#pragma unroll
        for (int v = 0; v < 8; ++v) {
            if ((v & m) == 0) {
                const int w = v | m;
                {
                    cplx a0{Xre[v], Xim[v]}, a1{Xre[w], Xim[w]};
                    cplx n0 = cadd(cmul(g.g00, a0), cmul(g.g01, a1));
                    cplx n1 = cadd(cmul(g.g10, a0), cmul(g.g11, a1));
                    Xre[v] = n0.re; Xim[v] = n0.im; Xre[w] = n1.re; Xim[w] = n1.im;
                }
                {
                    cplx a0{Yre[v], Yim[v]}, a1{Yre[w], Yim[w]};
                    cplx n0 = cadd(cmul(g.g00, a0), cmul(g.g01, a1));
                    cplx n1 = cadd(cmul(g.g10, a0), cmul(g.g11, a1));
                    Yre[v] = n0.re; Yim[v] = n0.im; Yre[w] = n1.re; Yim[w] = n1.im;
                }
            }
        }
    } else if constexpr (B == 8) {
#pragma unroll
        for (int v = 0; v < 8; ++v) {
            cplx a0{Xre[v], Xim[v]}, a1{Yre[v], Yim[v]};
            cplx n0 = cadd(cmul(g.g00, a0), cmul(g.g01, a1));
            cplx n1 = cadd(cmul(g.g10, a0), cmul(g.g11, a1));
            Xre[v] = n0.re; Xim[v] = n0.im; Yre[v] = n1.re; Yim[v] = n1.im;
        }
    } else {
        constexpr int mask = (B == 3) ? 16 : (1 << (B - 4));
        const bool hi = (lane & (unsigned)mask) != 0u;
#pragma unroll
        for (int v = 0; v < 8; ++v) {
            {
                cplx mine{Xre[v], Xim[v]};
                cplx part{__shfl_xor(Xre[v], mask, 32), __shfl_xor(Xim[v], mask, 32)};
                cplx a0 = hi ? part : mine;
                cplx a1 = hi ? mine : part;
                cplx glo = hi ? g.g10 : g.g00;
                cplx ghi = hi ? g.g11 : g.g01;
                cplx n = cadd(cmul(glo, a0), cmul(ghi, a1));
                Xre[v] = n.re; Xim[v] = n.im;
            }
            {
                cplx mine{Yre[v], Yim[v]};
                cplx part{__shfl_xor(Yre[v], mask, 32), __shfl_xor(Yim[v], mask, 32)};
                cplx a0 = hi ? part : mine;
                cplx a1 = hi ? mine : part;
                cplx glo = hi ? g.g10 : g.g00;
                cplx ghi = hi ? g.g11 : g.g01;
                cplx n = cadd(cmul(glo, a0), cmul(ghi, a1));
                Yre[v] = n.re; Yim[v] = n.im;
            }
        }
    }
}

// Broadcast S[R][lane&15] (row R of one 16x16 block held in C/D layout) to every lane.
template<int R>
__device__ __forceinline__ float rowb(const float* a, unsigned lane) {
    if constexpr (R < 8) return __shfl(a[R],     (int)(lane & 15u),        32);
    else                 return __shfl(a[R - 8], (int)((lane & 15u) + 16u), 32);
}

// B operand (4x16) for K-chunk starting at row K4: lanes 0-15 hold rows K4,K4+1;
// lanes 16-31 hold rows K4+2,K4+3 (mirrors the documented A 16x4 layout).
template<int K4>
__device__ __forceinline__ v2f makeB(const float* a, unsigned lane, bool hi) {
    float r0 = rowb<K4 + 0>(a, lane);
    float r1 = rowb<K4 + 1>(a, lane);
    float r2 = rowb<K4 + 2>(a, lane);
    float r3 = rowb<K4 + 3>(a, lane);
    v2f b;
    b.x = hi ? r2 : r0;
    b.y = hi ? r3 : r1;
    return b;
}

__global__ __launch_bounds__(256) void qconv9_kernel(const float* __restrict__ x,
                                                     const float* __restrict__ theta,
                                                     float* __restrict__ out)
{
    const unsigned lane = threadIdx.x & 31u;
    const unsigned wave = threadIdx.x >> 5;
    const unsigned p = blockIdx.x * 8u + wave;
    if (p >= NPATCH) return;                       // never taken (exact grid): EXEC stays all-1s

    const unsigned b   = p / (NOUT * NOUT);
    const unsigned rem = p % (NOUT * NOUT);
    const unsigned i   = rem / NOUT, j = rem % NOUT;

    // ---- per-patch data angles (RX half-angles) ----
    float dc[NQ], dsn[NQ];
#pragma unroll
    for (int q = 0; q < NQ; ++q) {
        float f = x[b * HW * HW + (i + q / 3) * HW + (j + q % 3)];
        float s, c;
        __sincosf(0.5f * f, &s, &c);
        dc[q] = c; dsn[q] = s;
    }

    // ---- shared parameterized gates: G_q = RZ(pi t2) * RY(pi t1) * RX(pi t0) ----
    gate2 G[NQ];
    const float PI = 3.14159265358979323846f;
#pragma unroll
    for (int q = 0; q < NQ; ++q) {
        float sa, ca, sb, cb, sg, cg;
        __sincosf(0.5f * PI * theta[q * 3 + 0], &sa, &ca);
        __sincosf(0.5f * PI * theta[q * 3 + 1], &sb, &cb);
        __sincosf(0.5f * PI * theta[q * 3 + 2], &sg, &cg);
        cplx rx00{ca, 0.f}, rx01{0.f, -sa}, rx10{0.f, -sa}, rx11{ca, 0.f};
        cplx m00 = cadd(cmul({cb, 0.f}, rx00), cmul({-sb, 0.f}, rx10));
        cplx m01 = cadd(cmul({cb, 0.f}, rx01), cmul({-sb, 0.f}, rx11));
        cplx m10 = cadd(cmul({sb, 0.f}, rx00), cmul({cb, 0.f}, rx10));
        cplx m11 = cadd(cmul({sb, 0.f}, rx01), cmul({cb, 0.f}, rx11));
        cplx e0{cg, -sg}, e1{cg, sg};
        G[q].g00 = cmul(e0, m00); G[q].g01 = cmul(e0, m01);
        G[q].g10 = cmul(e1, m10); G[q].g11 = cmul(e1, m11);
    }

    // ---- state |0...0>: flat index 0 = row 0, col 0, block X ----
    float sXre[8], sXim[8], sYre[8], sYim[8];
#pragma unroll
    for (int v = 0; v < 8; ++v) { sXre[v] = 0.f; sXim[v] = 0.f; sYre[v] = 0.f; sYim[v] = 0.f; }
    sXre[0] = (lane == 0u) ? 1.0f : 0.0f;

    // ---- data RX gates: qubit q acts on flat bit 8-q ----
    apply1q<8>(sXre, sXim, sYre, sYim, mk_rx(dc[0], dsn[0]), lane);
    apply1q<7>(sXre, sXim, sYre, sYim, mk_rx(dc[1], dsn[1]), lane);
    apply1q<6>(sXre, sXim, sYre, sYim, mk_rx(dc[2], dsn[2]), lane);
    apply1q<5>(sXre, sXim, sYre, sYim, mk_rx(dc[3], dsn[3]), lane);
    apply1q<4>(sXre, sXim, sYre, sYim, mk_rx(dc[4], dsn[4]), lane);
    apply1q<3>(sXre, sXim, sYre, sYim, mk_rx(dc[5], dsn[5]), lane);
    apply1q<2>(sXre, sXim, sYre, sYim, mk_rx(dc[6], dsn[6]), lane);
    apply1q<1>(sXre, sXim, sYre, sYim, mk_rx(dc[7], dsn[7]), lane);
    apply1q<0>(sXre, sXim, sYre, sYim, mk_rx(dc[8], dsn[8]), lane);

    // ---- all 9 CZs fused into one sign pass (all diagonal, commuting) ----
    // ring pairs (bit,bit+1) for bit=0..7  plus  (bit8,bit0)
#pragma unroll
    for (int v = 0; v < 8; ++v) {
        unsigned aX = (unsigned)v | ((lane >> 4) << 3) | ((lane & 15u) << 4);
        unsigned aY = aX | 256u;
        unsigned parX = __popc((aX & (aX >> 1)) & 0xFFu) + ((aX >> 8) & aX & 1u);
        unsigned parY = __popc((aY & (aY >> 1)) & 0xFFu) + ((aY >> 8) & aY & 1u);
        float fX = (parX & 1u) ? -1.f : 1.f;
        float fY = (parY & 1u) ? -1.f : 1.f;
        sXre[v] *= fX; sXim[v] *= fX;
        sYre[v] *= fY; sYim[v] *= fY;
    }

    // ---- parameterized gates on the 5 "column" qubits (bits 8..4) via shuffles ----
    apply1q<8>(sXre, sXim, sYre, sYim, G[0], lane);
    apply1q<7>(sXre, sXim, sYre, sYim, G[1], lane);
    apply1q<6>(sXre, sXim, sYre, sYim, G[2], lane);
    apply1q<5>(sXre, sXim, sYre, sYim, G[3], lane);
    apply1q<4>(sXre, sXim, sYre, sYim, G[4], lane);

    // ---- qubits 5..8 (row bits 3..0) fused: U = G5 (x) G6 (x) G7 (x) G8, S' = U @ S via WMMA ----
    // A operand (documented 16x4 layout): lane<16 holds U[lane][4k+{0,1}], lane>=16 holds U[lane-16][4k+{2,3}]
    const int  r   = (int)(lane & 15u);
    const bool hiL = lane >= 16u;
    v2f Are[4], Aim[4], Anim[4];
#pragma unroll
    for (int k = 0; k < 4; ++k) {
        cplx e[2];
#pragma unroll
        for (int v = 0; v < 2; ++v) {
            int col = 4 * k + (hiL ? 2 : 0) + v;
            cplx t0 = cmul(gpick(G[5], (r >> 3) & 1, (col >> 3) & 1),
                           gpick(G[6], (r >> 2) & 1, (col >> 2) & 1));
            cplx t1 = cmul(gpick(G[7], (r >> 1) & 1, (col >> 1) & 1),
                           gpick(G[8],  r       & 1,  col       & 1));
            e[v] = cmul(t0, t1);
        }
        Are[k]  = v2f{ e[0].re,  e[1].re};
        Aim[k]  = v2f{ e[0].im,  e[1].im};
        Anim[k] = v2f{-e[0].im, -e[1].im};   // f32 WMMA has no A-neg modifier: bake it in
    }

    v2f BXre[4], BXim[4], BYre[4], BYim[4];
    BXre[0] = makeB<0>(sXre, lane, hiL); BXre[1] = makeB<4>(sXre, lane, hiL);
    BXre[2] = makeB<8>(sXre, lane, hiL); BXre[3] = makeB<12>(sXre, lane, hiL);
    BXim[0] = makeB<0>(sXim, lane, hiL); BXim[1] = makeB<4>(sXim, lane, hiL);
    BXim[2] = makeB<8>(sXim, lane, hiL); BXim[3] = makeB<12>(sXim, lane, hiL);
    BYre[0] = makeB<0>(sYre, lane, hiL); BYre[1] = makeB<4>(sYre, lane, hiL);
    BYre[2] = makeB<8>(sYre, lane, hiL); BYre[3] = makeB<12>(sYre, lane, hiL);
    BYim[0] = makeB<0>(sYim, lane, hiL); BYim[1] = makeB<4>(sYim, lane, hiL);
    BYim[2] = makeB<8>(sYim, lane, hiL); BYim[3] = makeB<12>(sYim, lane, hiL);

    v8f nXre = {0, 0, 0, 0, 0, 0, 0, 0}, nXim = {0, 0, 0, 0, 0, 0, 0, 0};
    v8f nYre = {0, 0, 0, 0, 0, 0, 0, 0}, nYim = {0, 0, 0, 0, 0, 0, 0, 0};
#pragma unroll
    for (int k = 0; k < 4; ++k) nXre = wmma4(Are[k],  BXre[k], nXre);
#pragma unroll
    for (int k = 0; k < 4; ++k) nXre = wmma4(Anim[k], BXim[k], nXre);   // Re = Ur*Sr - Ui*Si
#pragma unroll
    for (int k = 0; k < 4; ++k) nXim = wmma4(Are[k],  BXim[k], nXim);
#pragma unroll
    for (int k = 0; k < 4; ++k) nXim = wmma4(Aim[k],  BXre[k], nXim);   // Im = Ur*Si + Ui*Sr
#pragma unroll
    for (int k = 0; k < 4; ++k) nYre = wmma4(Are[k],  BYre[k], nYre);
#pragma unroll
    for (int k = 0; k < 4; ++k) nYre = wmma4(Anim[k], BYim[k], nYre);
#pragma unroll
    for (int k = 0; k < 4; ++k) nYim = wmma4(Are[k],  BYim[k], nYim);
#pragma unroll
    for (int k = 0; k < 4; ++k) nYim = wmma4(Aim[k],  BYre[k], nYim);

    // ---- p1 = sum |amp|^2 over flat bit 8 = 1 (block Y), full-wave reduce ----
    float acc = 0.f;
#pragma unroll
    for (int v = 0; v < 8; ++v) acc += nYre[v] * nYre[v] + nYim[v] * nYim[v];
#pragma unroll
    for (int off = 16; off > 0; off >>= 1) acc += __shfl_xor(acc, off, 32);
    if (lane == 0u) out[p] = acc;
}

extern "C" void kernel_launch(void* const* d_in, const int* in_sizes, int n_in,
                              void* d_out, int out_size, void* d_ws, size_t ws_size,
                              hipStream_t stream) {
    const float* x     = (const float*)d_in[0];   // (16,28,28) f32
    const float* theta = (const float*)d_in[1];   // (1,27) f32
    float* out = (float*)d_out;                   // (16,26,26) f32
    dim3 grid(NPATCH / 8);                        // 1352 blocks * 8 waves = 10816 patches
    qconv9_kernel<<<grid, 256, 0, stream>>>(x, theta, out);
    (void)in_sizes; (void)n_in; (void)out_size; (void)d_ws; (void)ws_size;
}